// Net_84275848282731
// MI455X (gfx1250) — compile-verified
//
#include <hip/hip_runtime.h>
#include <hip/hip_bf16.h>

#define NEG_SLOPE_C 0.2f
#define BN_EPS_C 1e-5f

typedef float v2f __attribute__((ext_vector_type(2)));
typedef float v8f __attribute__((ext_vector_type(8)));

// ---------------- batch-norm column stats -> scale/shift ----------------
__global__ void bn_stats_k(const float* __restrict__ X, const float* __restrict__ gamma,
                           const float* __restrict__ beta, float* __restrict__ scale,
                           float* __restrict__ shift, int Nrows, int din) {
    int c = blockIdx.x;
    __shared__ float ssum[256], ssq[256];
    float s = 0.f, q = 0.f;
    for (int n = threadIdx.x; n < Nrows; n += blockDim.x) {
        float v = X[(size_t)n * din + c];
        s += v; q += v * v;
    }
    ssum[threadIdx.x] = s; ssq[threadIdx.x] = q;
    __syncthreads();
    for (int off = blockDim.x >> 1; off > 0; off >>= 1) {
        if (threadIdx.x < (unsigned)off) {
            ssum[threadIdx.x] += ssum[threadIdx.x + off];
            ssq[threadIdx.x]  += ssq[threadIdx.x + off];
        }
        __syncthreads();
    }
    if (threadIdx.x == 0) {
        float m = ssum[0] / (float)Nrows;
        float var = ssq[0] / (float)Nrows - m * m;
        float sc = gamma[c] * rsqrtf(var + BN_EPS_C);
        scale[c] = sc;
        shift[c] = beta[c] - m * sc;
    }
}

// ---------------- fused BN + GEMM via f32 WMMA (16x16x4) ----------------
// Out[n][0:OUT] = (X[n]*scale+shift) @ W + bias
// One 16-row tile per block; wave w owns col-tiles w and w+8 (A frag shared).
// All indices clamped pre-loop: EXEC stays all-1s, stores are guarded instead.
template <int DIN, int OUT>
__global__ __launch_bounds__(256) void gemm_bn_wmma(
    const float* __restrict__ X, const float* __restrict__ scale,
    const float* __restrict__ shift, const float* __restrict__ W,
    const float* __restrict__ bias, float* __restrict__ Out, int Nrows) {
    constexpr int NCT = (OUT + 15) / 16;       // total 16-col tiles
    constexpr int CTW = (NCT + 7) / 8;         // col tiles per wave
    const int row0 = blockIdx.x * 16;
    const int wave = threadIdx.x >> 5;
    const int lane = threadIdx.x & 31;
    const int lhalf = lane & 15;
    const bool hi = lane >= 16;
    if (wave >= NCT) return;                   // wave-uniform

    int row = row0 + lhalf;
    if (row >= Nrows) row = Nrows - 1;         // clamp: garbage rows never stored
    const int kofs = hi ? 2 : 0;               // this lane holds K = {kofs, kofs+1} (mod 4)
    const float* Xrow = X + (size_t)row * DIN + kofs;
    const float* sp = scale + kofs;
    const float* tp = shift + kofs;

    const float* Wc[CTW];
    int  colA[CTW];
    bool colOk[CTW];
    #pragma unroll
    for (int t = 0; t < CTW; ++t) {
        int c = (wave + t * 8) * 16 + lhalf;
        colOk[t] = c < OUT;
        colA[t] = colOk[t] ? c : (OUT - 1);    // clamp: garbage cols never stored
        Wc[t] = W + colA[t] + (size_t)kofs * OUT;
    }

    v8f acc[CTW];
    #pragma unroll
    for (int t = 0; t < CTW; ++t) acc[t] = (v8f){};

    #pragma unroll 8
    for (int k0 = 0; k0 < DIN; k0 += 4) {
        v2f xv = *(const v2f*)(Xrow + k0);     // X[row][k0+kofs .. +1]
        v2f sv = *(const v2f*)(sp + k0);
        v2f tv = *(const v2f*)(tp + k0);
        v2f a;
        a.x = xv.x * sv.x + tv.x;              // fused batch-norm on A
        a.y = xv.y * sv.y + tv.y;
        #pragma unroll
        for (int t = 0; t < CTW; ++t) {
            v2f b;
            b.x = Wc[t][(size_t)k0 * OUT];         // W[k0+kofs][col]
            b.y = Wc[t][(size_t)(k0 + 1) * OUT];   // W[k0+kofs+1][col]
            acc[t] = __builtin_amdgcn_wmma_f32_16x16x4_f32(
                false, a, false, b, (short)0, acc[t], false, false);
        }
    }

    const int rbase = row0 + (hi ? 8 : 0);
    #pragma unroll
    for (int t = 0; t < CTW; ++t) {
        if (!colOk[t]) continue;
        const float bv = bias ? bias[colA[t]] : 0.f;
        #pragma unroll
        for (int i = 0; i < 8; ++i) {
            int r = rbase + i;
            if (r < Nrows) Out[(size_t)r * OUT + colA[t]] = acc[t][i] + bv;
        }
    }
}

// ---------------- per-node attention dot products ----------------
template <int HEADS, int DOUT>
__global__ void attn_scores_k(const float* __restrict__ H, const float* __restrict__ Asrc,
                              const float* __restrict__ Adst, float* __restrict__ es,
                              float* __restrict__ ed, int Nrows) {
    int idx = blockIdx.x * blockDim.x + threadIdx.x;
    if (idx >= Nrows * HEADS) return;
    int n = idx / HEADS, h = idx % HEADS;
    const float* hp = H + (size_t)n * HEADS * DOUT + h * DOUT;
    const float* as = Asrc + h * DOUT;
    const float* ad = Adst + h * DOUT;
    float s = 0.f, d = 0.f;
    #pragma unroll 8
    for (int i = 0; i < DOUT; ++i) { float v = hp[i]; s += v * as[i]; d += v * ad[i]; }
    es[idx] = s; ed[idx] = d;
}

// ---------------- ordered float <-> uint encoding for atomicMax ----------------
__device__ __forceinline__ unsigned enc_f(float f) {
    unsigned u = __float_as_uint(f);
    return (u >> 31) ? ~u : (u | 0x80000000u);
}
__device__ __forceinline__ float dec_f(unsigned e) {
    return (e >> 31) ? __uint_as_float(e & 0x7FFFFFFFu) : __uint_as_float(~e);
}

template <int HEADS>
__global__ void edge_max_k(const int* __restrict__ src, const int* __restrict__ dst,
                           const int* __restrict__ ecls, int cls,
                           const float* __restrict__ es, const float* __restrict__ ed,
                           unsigned* __restrict__ emaxU, int E_) {
    int e = blockIdx.x * blockDim.x + threadIdx.x;
    if (e >= E_ || ecls[e] != cls) return;
    int s = src[e], d = dst[e];
    #pragma unroll
    for (int h = 0; h < HEADS; ++h) {
        float v = es[s * HEADS + h] + ed[d * HEADS + h];
        v = v > 0.f ? v : v * NEG_SLOPE_C;                 // leaky_relu
        atomicMax(&emaxU[d * HEADS + h], enc_f(v));
    }
}

// wave-per-edge: softmax numerator + scatter message (atomics stay L2-resident)
template <int HEADS, int DOUT>
__global__ void edge_sum_k(const int* __restrict__ src, const int* __restrict__ dst,
                           const int* __restrict__ ecls, int cls,
                           const float* __restrict__ es, const float* __restrict__ ed,
                           const unsigned* __restrict__ emaxU, const float* __restrict__ H,
                           float* __restrict__ denom, float* __restrict__ msg, int E_) {
    int wid = (int)((blockIdx.x * (size_t)blockDim.x + threadIdx.x) >> 5);
    int lane = threadIdx.x & 31;
    if (wid >= E_ || ecls[wid] != cls) return;
    int s = src[wid], d = dst[wid];
    constexpr int D = HEADS * DOUT;
    #pragma unroll
    for (int h = 0; h < HEADS; ++h) {
        float v = es[s * HEADS + h] + ed[d * HEADS + h];
        v = v > 0.f ? v : v * NEG_SLOPE_C;
        unsigned mu = emaxU[d * HEADS + h];
        float m = (mu == 0u) ? 0.f : dec_f(mu);
        float a = expf(v - m);
        if (lane == 0) atomicAdd(&denom[d * HEADS + h], a);
        #pragma unroll
        for (int i = lane; i < DOUT; i += 32) {
            atomicAdd(&msg[(size_t)d * D + h * DOUT + i],
                      a * H[(size_t)s * D + h * DOUT + i]);
        }
    }
}

// x0 += msg/denom + bias
__global__ void finalize_k(const float* __restrict__ denom, const float* __restrict__ msg,
                           const float* __restrict__ cbias, float* __restrict__ x0,
                           int Nrows, int heads, int dout) {
    int idx = blockIdx.x * blockDim.x + threadIdx.x;
    int D = heads * dout;
    if (idx >= Nrows * D) return;
    int n = idx / D, c = idx % D;
    int h = c / dout;
    float den = denom[n * heads + h];
    den = den > 0.f ? den : 1.0f;
    x0[idx] += msg[idx] / den + cbias[c];
}

__global__ void relu_k(float* __restrict__ x, int n) {
    int i = blockIdx.x * blockDim.x + threadIdx.x;
    if (i < n) x[i] = fmaxf(x[i], 0.f);
}

// ---------------- host launcher ----------------
extern "C" void kernel_launch(void* const* d_in, const int* in_sizes, int n_in,
                              void* d_out, int out_size, void* d_ws, size_t ws_size,
                              hipStream_t stream) {
    const int F_IN = 64;
    const int N = in_sizes[0] / F_IN;
    const int E = in_sizes[1] / 2;

    const float* x_in = (const float*)d_in[0];
    const int* eidx   = (const int*)d_in[1];
    const int* src    = eidx;
    const int* dst    = eidx + E;
    const int* eattr  = (const int*)d_in[2];

    const int dins[3]   = {64, 256, 256};
    const int headsA[3] = {4, 4, 1};
    const int doutA[3]  = {64, 64, 2};

    // workspace layout (floats)
    float* ws = (float*)d_ws;
    size_t off = 0;
    float*    buf0  = ws + off; off += (size_t)N * 256;
    float*    buf1  = ws + off; off += (size_t)N * 256;
    float*    hbuf  = ws + off; off += (size_t)N * 256;
    float*    msg   = ws + off; off += (size_t)N * 256;
    float*    es    = ws + off; off += (size_t)N * 4;
    float*    ed    = ws + off; off += (size_t)N * 4;
    float*    denom = ws + off; off += (size_t)N * 4;
    unsigned* emaxU = (unsigned*)(ws + off); off += (size_t)N * 4;
    float*    scale = ws + off; off += 256;
    float*    shift = ws + off; off += 256;

    float* outf = (float*)d_out;
    const int rowTiles = (N + 15) / 16;

    for (int l = 0; l < 3; ++l) {
        const int din   = dins[l];
        const int heads = headsA[l];
        const int dout  = doutA[l];
        const int D     = heads * dout;

        const float* W   = (const float*)d_in[3 + 8 * l];
        const float* As  = (const float*)d_in[4 + 8 * l];
        const float* Ad  = (const float*)d_in[5 + 8 * l];
        const float* cb  = (const float*)d_in[6 + 8 * l];
        const float* skW = (const float*)d_in[7 + 8 * l];
        const float* skb = (const float*)d_in[8 + 8 * l];
        const float* bng = (const float*)d_in[9 + 8 * l];
        const float* bnb = (const float*)d_in[10 + 8 * l];

        const float* cur = (l == 0) ? x_in : ((l == 1) ? buf0 : buf1);
        float* x0        = (l == 0) ? buf0 : ((l == 1) ? buf1 : outf);

        // 1. BN column stats
        bn_stats_k<<<din, 256, 0, stream>>>(cur, bng, bnb, scale, shift, N, din);

        // 2. skip connection: x0 = bn(x) @ skW + skb
        if (l == 0)      gemm_bn_wmma<64, 256><<<rowTiles, 256, 0, stream>>>(cur, scale, shift, skW, skb, x0, N);
        else if (l == 1) gemm_bn_wmma<256, 256><<<rowTiles, 256, 0, stream>>>(cur, scale, shift, skW, skb, x0, N);
        else             gemm_bn_wmma<256, 2><<<rowTiles, 256, 0, stream>>>(cur, scale, shift, skW, skb, x0, N);

        // 3. per edge-class GAT conv, accumulated into x0
        for (int j = 0; j < 4; ++j) {
            const float* Wj = W + (size_t)j * din * D;
            if (l == 0)      gemm_bn_wmma<64, 256><<<rowTiles, 256, 0, stream>>>(cur, scale, shift, Wj, nullptr, hbuf, N);
            else if (l == 1) gemm_bn_wmma<256, 256><<<rowTiles, 256, 0, stream>>>(cur, scale, shift, Wj, nullptr, hbuf, N);
            else             gemm_bn_wmma<256, 2><<<rowTiles, 256, 0, stream>>>(cur, scale, shift, Wj, nullptr, hbuf, N);

            if (l < 2) attn_scores_k<4, 64><<<(N * heads + 255) / 256, 256, 0, stream>>>(
                           hbuf, As + (size_t)j * D, Ad + (size_t)j * D, es, ed, N);
            else       attn_scores_k<1, 2><<<(N * heads + 255) / 256, 256, 0, stream>>>(
                           hbuf, As + (size_t)j * D, Ad + (size_t)j * D, es, ed, N);

            hipMemsetAsync(emaxU, 0, (size_t)N * heads * sizeof(unsigned), stream);
            hipMemsetAsync(denom, 0, (size_t)N * heads * sizeof(float), stream);
            hipMemsetAsync(msg,   0, (size_t)N * D * sizeof(float), stream);

            if (l < 2) edge_max_k<4><<<(E + 255) / 256, 256, 0, stream>>>(src, dst, eattr, j, es, ed, emaxU, E);
            else       edge_max_k<1><<<(E + 255) / 256, 256, 0, stream>>>(src, dst, eattr, j, es, ed, emaxU, E);

            size_t esum_blocks = ((size_t)E * 32 + 255) / 256;
            if (l < 2) edge_sum_k<4, 64><<<(unsigned)esum_blocks, 256, 0, stream>>>(
                           src, dst, eattr, j, es, ed, emaxU, hbuf, denom, msg, E);
            else       edge_sum_k<1, 2><<<(unsigned)esum_blocks, 256, 0, stream>>>(
                           src, dst, eattr, j, es, ed, emaxU, hbuf, denom, msg, E);

            finalize_k<<<((size_t)N * D + 255) / 256, 256, 0, stream>>>(
                denom, msg, cb + (size_t)j * D, x0, N, heads, dout);
        }

        // 4. ReLU -> becomes next layer's input (or final output)
        relu_k<<<((size_t)N * D + 255) / 256, 256, 0, stream>>>(x0, N * D);
    }
}